// ContrastiveLoss_25675314495874
// MI455X (gfx1250) — compile-verified
//
#include <hip/hip_runtime.h>
#include <math.h>

// NT-Xent (SimCLR) contrastive loss, fused X·X^T + exp + row-sum on gfx1250.
// Uses V_WMMA_F32_16X16X4_F32 to keep full fp32 precision for the logits.

typedef float v2f __attribute__((ext_vector_type(2)));
typedef float v8f __attribute__((ext_vector_type(8)));

#define B_ROWS 4096
#define DIM    128
#define N_TOT  8192          // 2B
#define INV_T  2.0f          // 1 / 0.5

__device__ __forceinline__ const float* row_ptr(const float* o1, const float* o2, int r) {
    return (r < B_ROWS) ? (o1 + r * DIM) : (o2 + (r - B_ROWS) * DIM);
}

__global__ __launch_bounds__(256)
void ntxent_rowsum_kernel(const float* __restrict__ out1,
                          const float* __restrict__ out2,
                          float* __restrict__ block_partials) {
    __shared__ float As[32 * DIM];   // 32 rows x 128 cols of X (this block's row tile)
    __shared__ float rowSum[32];
    __shared__ float rowLoss[32];

    const int tid  = threadIdx.x;
    const int lane = tid & 31;
    const int wave = tid >> 5;
    const int r0   = blockIdx.x * 32;

    // Cooperative stage of the 32-row A tile into LDS (float4 granules).
    for (int q = tid; q < (32 * DIM) / 4; q += 256) {
        const int row = q >> 5;      // 32 float4 per row
        const int c4  = q & 31;
        const float4* src = (const float4*)(row_ptr(out1, out2, r0 + row) + c4 * 4);
        ((float4*)As)[row * 32 + c4] = *src;
    }
    if (tid < 32) rowSum[tid] = 0.0f;
    __syncthreads();

    const int half = lane >> 4;      // K sub-slot: lanes 0-15 hold k0,k1; 16-31 hold k2,k3
    const int l16  = lane & 15;
    const int koff = half * 2;

    v8f sum0 = {};                   // exp row-partials, rows r0..r0+15
    v8f sum1 = {};                   // rows r0+16..r0+31

    const float* a0p = As + l16 * DIM + koff;
    const float* a1p = As + (16 + l16) * DIM + koff;

    // 8 waves stripe the 512 column tiles.
    for (int t = wave; t < N_TOT / 16; t += 8) {
        const int c0 = t * 16;
        const float* bptr = row_ptr(out1, out2, c0 + l16) + koff;

        // Prefetch next column tile's row for this lane.
        const int tn = t + 8;
        if (tn < N_TOT / 16)
            __builtin_prefetch(row_ptr(out1, out2, tn * 16 + l16), 0, 0);

        v8f acc0 = {};
        v8f acc1 = {};
        #pragma unroll 8
        for (int s = 0; s < DIM / 4; ++s) {
            v2f bf = *(const v2f*)(bptr + 4 * s);       // B frag: column c0+l16, K slice
            v2f a0 = *(const v2f*)(a0p + 4 * s);        // A frag rows r0..r0+15
            v2f a1 = *(const v2f*)(a1p + 4 * s);        // A frag rows r0+16..r0+31
            acc0 = __builtin_amdgcn_wmma_f32_16x16x4_f32(false, a0, false, bf,
                                                         (short)0, acc0, false, false);
            acc1 = __builtin_amdgcn_wmma_f32_16x16x4_f32(false, a1, false, bf,
                                                         (short)0, acc1, false, false);
        }

        // exp(sim/T), excluding the diagonal self-similarity.
        // C/D layout: lane l16 = column n; VGPR v = row (v + 8*half).
        #pragma unroll
        for (int v = 0; v < 8; ++v) {
            const int col = c0 + l16;
            const int m   = v + 8 * half;
            float e0 = __expf(INV_T * acc0[v]);
            if (r0 + m == col) e0 = 0.0f;
            sum0[v] += e0;
            float e1 = __expf(INV_T * acc1[v]);
            if (r0 + 16 + m == col) e1 = 0.0f;
            sum1[v] += e1;
        }
    }

    // Reduce each row partial across its 16 owning lanes, merge waves via LDS atomics.
    #pragma unroll
    for (int v = 0; v < 8; ++v) {
        float s0 = sum0[v];
        float s1 = sum1[v];
        for (int off = 1; off < 16; off <<= 1) {
            s0 += __shfl_xor(s0, off, 32);
            s1 += __shfl_xor(s1, off, 32);
        }
        if (l16 == 0) {
            atomicAdd(&rowSum[v + 8 * half], s0);        // ds_add_f32
            atomicAdd(&rowSum[16 + v + 8 * half], s1);
        }
    }
    __syncthreads();

    // Per-row loss: log(denom) - dot(out1_i, out2_i)/T  (positive pair term).
    if (tid < 32) {
        const int gr  = r0 + tid;
        const int idx = (gr < B_ROWS) ? gr : gr - B_ROWS;
        const float* p1 = out1 + idx * DIM;
        const float* p2 = out2 + idx * DIM;
        float dot = 0.0f;
        #pragma unroll 16
        for (int k = 0; k < DIM; ++k) dot = fmaf(p1[k], p2[k], dot);
        rowLoss[tid] = __logf(rowSum[tid]) - INV_T * dot;
    }
    __syncthreads();

    if (tid == 0) {
        float s = 0.0f;                                  // fixed-order: deterministic
        #pragma unroll
        for (int i = 0; i < 32; ++i) s += rowLoss[i];
        block_partials[blockIdx.x] = s;
    }
}

__global__ __launch_bounds__(256)
void ntxent_final_reduce_kernel(const float* __restrict__ block_partials,
                                float* __restrict__ out) {
    __shared__ float sm[256];
    const int t = threadIdx.x;
    sm[t] = block_partials[t];
    __syncthreads();
    for (int s = 128; s > 0; s >>= 1) {                  // fixed tree order
        if (t < s) sm[t] += sm[t + s];
        __syncthreads();
    }
    if (t == 0) out[0] = sm[0] * (1.0f / (float)N_TOT);
}

extern "C" void kernel_launch(void* const* d_in, const int* in_sizes, int n_in,
                              void* d_out, int out_size, void* d_ws, size_t ws_size,
                              hipStream_t stream) {
    const float* out1 = (const float*)d_in[0];
    const float* out2 = (const float*)d_in[1];
    float* loss       = (float*)d_out;
    float* partials   = (float*)d_ws;    // 256 floats of scratch

    (void)in_sizes; (void)n_in; (void)out_size; (void)ws_size;

    ntxent_rowsum_kernel<<<N_TOT / 32, 256, 0, stream>>>(out1, out2, partials);
    ntxent_final_reduce_kernel<<<1, 256, 0, stream>>>(partials, loss);
}